// TotalVariationDenoising_56573309223713
// MI455X (gfx1250) — compile-verified
//
#include <hip/hip_runtime.h>
#include <stdint.h>

// -----------------------------------------------------------------------------
// TV denoising (Chambolle primal-dual), 100 iterations, B=8, C=1, H=W=512 fp32.
//
// MI455X reasoning: working set ~34 MB -> L2-resident (192 MB L2); bound by
// launch latency + L2 bandwidth. This version fuses TWO primal-dual iterations
// per kernel (temporal blocking with a +1 halo):
//   x1 = f(noisy, u0)           (x1 does not depend on x0)
//   u1 = f(u0, grad(x0))
//   x2 = f(noisy, u1)
//   u2 = f(u1, grad(x1))
// => 50 launches instead of 100, and ~half the global (L2) traffic; the extra
// halo recompute lives entirely in LDS/VALU which are idle anyway.
// Tiles are staged with gfx1250 async-to-LDS loads (ASYNCcnt path).
// -----------------------------------------------------------------------------

#define HH   512
#define WW   512
#define BB   8
#define NPLANE ((size_t)BB * HH * WW)   // 2,097,152 elements per plane
#define TAU_ 0.01f
#define LAM_ 0.1f
#define EPS_ 1e-10f
#define STEPS_ 50                       // 50 launches x 2 iterations = 100

#if defined(__has_builtin)
#if __has_builtin(__builtin_amdgcn_global_load_async_to_lds_b32)
#define TV_HAVE_ASYNC 1
#endif
#endif

typedef __attribute__((address_space(1))) int* gint_p;
typedef __attribute__((address_space(3))) int* lint_p;

__device__ __forceinline__ void tv_stage(const float* g, float* l) {
#ifdef TV_HAVE_ASYNC
  __builtin_amdgcn_global_load_async_to_lds_b32((gint_p)g, (lint_p)l, 0, 0);
#else
  *l = *g;
#endif
}

__device__ __forceinline__ void tv_stage_wait() {
#ifdef TV_HAVE_ASYNC
#if __has_builtin(__builtin_amdgcn_s_wait_asynccnt)
  __builtin_amdgcn_s_wait_asynccnt(0);
#else
  asm volatile("s_wait_asynccnt 0x0" ::: "memory");
#endif
#endif
}

__device__ __forceinline__ float tv_norm(float v) {
  return v / (fabsf(v) + EPS_);
}
__device__ __forceinline__ float tv_xupd(float nz, float div) {
  float xn = nz + TAU_ * div;
  return (xn + TAU_ * nz) / (1.0f + TAU_);
}

// Two fused primal-dual iterations per launch.
// u layout: [ux planes for all batches][uy planes for all batches].
// LDS tile maps: s*(34x34) index [rq][cq] <-> global (i0+rq-1, j0+cq-1)
//                s*(33x33) index [r][c]   <-> global (i0+r,    j0+c)
__global__ __launch_bounds__(256)
void tv_iter2_kernel(const float* __restrict__ noisy,
                     const float* __restrict__ x,
                     const float* __restrict__ u,
                     float* __restrict__ xo,
                     float* __restrict__ uo,
                     int u_is_zero)
{
  __shared__ float sx0 [34 * 34];   // x0  on [-1..32]^2 (clamped halo)
  __shared__ float su0x[34 * 34];   // u0x on [-1..32]^2
  __shared__ float su0y[34 * 34];   // u0y on [-1..32]^2
  __shared__ float snz [33 * 33];   // noisy on [0..32]^2
  __shared__ float sx1 [33 * 33];   // x1  on [0..32]^2
  __shared__ float su1x[34 * 34];   // u1x on [-1..32]^2 (edge cells unused)
  __shared__ float su1y[34 * 34];   // u1y on [-1..32]^2

  const int b   = blockIdx.z;
  const int i0  = blockIdx.y * 32;
  const int j0  = blockIdx.x * 32;
  const int tid = threadIdx.x;

  const size_t base = (size_t)b * (HH * WW);
  const float* xb  = x + base;
  const float* uxb = u + base;
  const float* uyb = u + NPLANE + base;
  const float* nzb = noisy + base;

  // ---- stage x0 (+ u0) on the 34x34 halo grid, clamped at image edges ------
#pragma unroll
  for (int t = 0; t < 5; ++t) {
    int idx = tid + t * 256;
    if (idx < 34 * 34) {
      int r = idx / 34, c = idx - r * 34;
      int gi = i0 + r - 1; gi = gi < 0 ? 0 : (gi > HH - 1 ? HH - 1 : gi);
      int gj = j0 + c - 1; gj = gj < 0 ? 0 : (gj > WW - 1 ? WW - 1 : gj);
      size_t o = (size_t)gi * WW + gj;
      tv_stage(xb + o, &sx0[idx]);
      if (!u_is_zero) {
        tv_stage(uxb + o, &su0x[idx]);
        tv_stage(uyb + o, &su0y[idx]);
      }
    }
  }
  // ---- stage noisy on the 33x33 grid ---------------------------------------
#pragma unroll
  for (int t = 0; t < 5; ++t) {
    int idx = tid + t * 256;
    if (idx < 33 * 33) {
      int r = idx / 33, c = idx - r * 33;
      int gi = i0 + r; if (gi > HH - 1) gi = HH - 1;
      int gj = j0 + c; if (gj > WW - 1) gj = WW - 1;
      tv_stage(nzb + (size_t)gi * WW + gj, &snz[idx]);
    }
  }
  tv_stage_wait();
  __syncthreads();

  const float K = TAU_ / LAM_;

  // ---- phase 1a: u1 = norm(u0 + K*grad(x0)) on the 34x34 grid --------------
#pragma unroll
  for (int t = 0; t < 5; ++t) {
    int idx = tid + t * 256;
    if (idx < 34 * 34) {
      int rq = idx / 34, cq = idx - rq * 34;
      int qi = i0 + rq - 1, qj = j0 + cq - 1;
      float x0c = sx0[idx];
      float gx0 = (cq < 33 && qj < WW - 1) ? (x0c - sx0[idx + 1])  : 0.0f;
      float gy0 = (rq < 33 && qi < HH - 1) ? (x0c - sx0[idx + 34]) : 0.0f;
      float u0xc = u_is_zero ? 0.0f : su0x[idx];
      float u0yc = u_is_zero ? 0.0f : su0y[idx];
      su1x[idx] = tv_norm(u0xc + K * gx0);
      su1y[idx] = tv_norm(u0yc + K * gy0);
    }
  }
  // ---- phase 1b: x1 = f(noisy, div(u0)) on the 33x33 grid ------------------
#pragma unroll
  for (int t = 0; t < 5; ++t) {
    int idx = tid + t * 256;
    if (idx < 33 * 33) {
      int r = idx / 33, c = idx - r * 33;
      int i = i0 + r, j = j0 + c;
      float div0 = 0.0f;
      if (!u_is_zero) {
        float dx = (j > 0) ? (su0x[(r + 1) * 34 + c + 1] - su0x[(r + 1) * 34 + c]) : 0.0f;
        float dy = (i > 0) ? (su0y[(r + 1) * 34 + c + 1] - su0y[r * 34 + c + 1])   : 0.0f;
        div0 = dx + dy;
      }
      sx1[idx] = tv_xupd(snz[idx], div0);
    }
  }
  __syncthreads();

  // ---- phase 2: x2, u2 on the 32x32 output tile ----------------------------
  const int c  = tid & 31;
  const int r0 = tid >> 5;
#pragma unroll
  for (int k = 0; k < 4; ++k) {
    int r = r0 + (k << 3);
    int i = i0 + r, j = j0 + c;

    float u1xc = su1x[(r + 1) * 34 + c + 1];   // u1x(i, j)
    float u1xm = su1x[(r + 1) * 34 + c];       // u1x(i, j-1)
    float u1yc = su1y[(r + 1) * 34 + c + 1];   // u1y(i, j)
    float u1ym = su1y[r * 34 + c + 1];         // u1y(i-1, j)

    float div1 = ((j > 0) ? (u1xc - u1xm) : 0.0f)
               + ((i > 0) ? (u1yc - u1ym) : 0.0f);

    float x2 = tv_xupd(snz[r * 33 + c], div1);

    float x1c = sx1[r * 33 + c];
    float gx1 = (j < WW - 1) ? (x1c - sx1[r * 33 + c + 1])   : 0.0f;
    float gy1 = (i < HH - 1) ? (x1c - sx1[(r + 1) * 33 + c]) : 0.0f;

    size_t off = base + (size_t)i * WW + j;
    xo[off]          = x2;
    uo[off]          = tv_norm(u1xc + K * gx1);
    uo[NPLANE + off] = tv_norm(u1yc + K * gy1);
  }
}

extern "C" void kernel_launch(void* const* d_in, const int* in_sizes, int n_in,
                              void* d_out, int out_size, void* d_ws, size_t ws_size,
                              hipStream_t stream) {
  (void)in_sizes; (void)n_in; (void)out_size; (void)ws_size;

  const float* noisy = (const float*)d_in[0];
  float* out = (float*)d_out;

  // Workspace layout (40 MB total):
  //   xA : NPLANE floats   (x ping buffer; d_out is the pong buffer)
  //   uA : 2*NPLANE floats
  //   uB : 2*NPLANE floats
  float* xA = (float*)d_ws;
  float* uA = xA + NPLANE;
  float* uB = uA + 2 * NPLANE;

  dim3 grid(WW / 32, HH / 32, BB);
  dim3 block(256);

  for (int s = 1; s <= STEPS_; ++s) {
    const float* xsrc = (s == 1) ? noisy : ((s % 2 == 0) ? xA : out);
    float*       xdst = (s % 2 == 0) ? out : xA;   // s=50 (even) -> d_out
    const float* usrc = (s % 2 == 1) ? uA : uB;    // unused when s==1
    float*       udst = (s % 2 == 1) ? uB : uA;

    tv_iter2_kernel<<<grid, block, 0, stream>>>(noisy, xsrc, usrc, xdst, udst,
                                                (s == 1) ? 1 : 0);
  }
}